// QFNNLayer_14963666059658
// MI455X (gfx1250) — compile-verified
//
#include <hip/hip_runtime.h>

// ---------------------------------------------------------------------------
// QFNN layer for MI455X (gfx1250, wave32).
// similarity via V_WMMA_F32_16X16X4_F32: the K=4 contraction encodes
//   sim[i,j] = x_i x_j + y_i y_j + ALPHA*(x_i y_j - y_i x_j)
// with A_i = (x_i, y_i, ALPHA*x_i, -ALPHA*y_i), B_j = (x_j, y_j, y_j, x_j).
// Store-bandwidth-bound: 268 MB attention output -> ~11.5 us at 23.3 TB/s.
// v_rcp_f32 gate keeps VALU under the store drain rate; attention streams
// out with non-temporal stores through a loop-carried row-base pointer
// (stores use the 24-bit immediate offset for the 8 row strides); column
// tile loads are software-pipelined one deep.
// ---------------------------------------------------------------------------

typedef float v2f __attribute__((ext_vector_type(2)));
typedef float v8f __attribute__((ext_vector_type(8)));

#define BB 4
#define NN 4096
#define NTILES (NN / 16)

constexpr float EPSILON     = 1e-6f;
constexpr float ALPHA       = 0.1f;
constexpr float DT          = 0.1f;
constexpr float RADIUS_RATE = 0.01f;
constexpr float R_MIN       = 0.1f;
constexpr float R_MAX       = 2.0f;
// sigmoid(BETA*(DT-1)) = 1/(1+exp(0.9)) — compile-time constant
constexpr float SKIP_W      = 0.28905049737499605f;

__device__ __forceinline__ float gate_att(float sim) {
    // gate = 1/(eps+|sim|) > 0.5  <=>  |sim| < 2 - eps
    float a = fabsf(sim);
    float att = sim * __builtin_amdgcn_rcpf(EPSILON + a);   // v_rcp_f32 + v_mul
    return (a < 2.0f - EPSILON) ? att : 0.0f;
}

// Pass 1: attention (stored), per-row context, attn_sum, mid_states.
__global__ __launch_bounds__(256) void qfnn_pass1(
    const float* __restrict__ states,   // [B,N,2]
    float* __restrict__ att_out,        // [B,N,N]
    float* __restrict__ ws_ctx,         // [B,N,2]
    float* __restrict__ ws_sum,         // [B,N]
    float* __restrict__ ws_mid)         // [B,N,2]
{
    const int lane = threadIdx.x & 31;
    const int wave = blockIdx.x * 8 + (threadIdx.x >> 5);   // 0..1023
    const int b    = wave >> 8;                             // batch
    const int rt   = wave & 255;                            // 16-row tile
    const int m    = lane & 15;
    const int hi   = lane >> 4;

    const float* sb = states + (size_t)b * NN * 2;
    float* attb     = att_out + (size_t)b * NN * NN;

    // A fragment (16x4 f32): lanes 0-15 hold K=0,1 ; lanes 16-31 hold K=2,3.
    const int rowA = rt * 16 + m;
    v2f sA = *(const v2f*)(sb + rowA * 2);
    v2f A;
    A.x = hi ? (ALPHA * sA.x) : sA.x;    // K=0: x_i   | K=2: alpha*x_i
    A.y = hi ? (-ALPHA * sA.y) : sA.y;   // K=1: y_i   | K=3: -alpha*y_i

    float asum[8] = {}, acx[8] = {}, acy[8] = {};

    // Loop-carried store pointer: row base for this lane's v=0 row; the 8 row
    // strides (v*NN floats = v*16KB) go into the store immediate offset.
    float* prow = attb + (size_t)(rt * 16 + 8 * hi) * NN + m;

    // one-deep software pipeline on the column-tile states
    v2f t = *(const v2f*)(sb + ((size_t)m) * 2);            // tile 0
    for (int ct = 0; ct < NTILES; ++ct) {
        const int jn = (((ct + 1) & (NTILES - 1)) << 4) + m;
        v2f tn = *(const v2f*)(sb + (size_t)jn * 2);        // prefetch ct+1

        // B fragment (4x16 f32): lanes 0-15 hold K=0,1 ; lanes 16-31 K=2,3.
        v2f Bv;
        Bv.x = hi ? t.y : t.x;                              // K=0: x_j | K=2: y_j
        Bv.y = hi ? t.x : t.y;                              // K=1: y_j | K=3: x_j

        v8f c = {};
        v8f d = __builtin_amdgcn_wmma_f32_16x16x4_f32(
            /*neg_a=*/false, A, /*neg_b=*/false, Bv,
            /*c_mod=*/(short)0, c, /*reuse_a=*/false, /*reuse_b=*/false);

#pragma unroll
        for (int v = 0; v < 8; ++v) {
            float att = gate_att(d[v]);                     // D layout: M = v + 8*hi
            __builtin_nontemporal_store(att, prow + (size_t)v * NN);
            asum[v] += att;
            acx[v]  += att * t.x;                           // context partial (col j)
            acy[v]  += att * t.y;
        }
        prow += 16;                                         // next 16-column tile
        t = tn;
    }

    // Row reductions across each 16-lane half (masks stay within halves).
#pragma unroll
    for (int v = 0; v < 8; ++v) {
#pragma unroll
        for (int mask = 1; mask < 16; mask <<= 1) {
            asum[v] += __shfl_xor(asum[v], mask, 32);
            acx[v]  += __shfl_xor(acx[v],  mask, 32);
            acy[v]  += __shfl_xor(acy[v],  mask, 32);
        }
    }

    if (m == 0) {
#pragma unroll
        for (int v = 0; v < 8; ++v) {
            const int row  = rt * 16 + v + 8 * hi;
            const size_t i = (size_t)b * NN + row;
            ws_ctx[i * 2]     = acx[v];
            ws_ctx[i * 2 + 1] = acy[v];
            ws_sum[i]         = asum[v];
            v2f st = *(const v2f*)(sb + row * 2);
            ws_mid[i * 2]     = st.x + DT * acx[v];
            ws_mid[i * 2 + 1] = st.y + DT * acy[v];
        }
    }
}

// Pass 2: attention at mid_states (context only) + Heun finalize + radius.
__global__ __launch_bounds__(256) void qfnn_pass2(
    const float* __restrict__ states,   // [B,N,2]
    const float* __restrict__ mid,      // [B,N,2]
    const float* __restrict__ ws_ctx,   // [B,N,2]
    const float* __restrict__ ws_sum,   // [B,N]
    float* __restrict__ out_states)     // [B,N,2]
{
    const int lane = threadIdx.x & 31;
    const int wave = blockIdx.x * 8 + (threadIdx.x >> 5);
    const int b    = wave >> 8;
    const int rt   = wave & 255;
    const int m    = lane & 15;
    const int hi   = lane >> 4;

    const float* mb = mid + (size_t)b * NN * 2;
    const float* sb = states + (size_t)b * NN * 2;

    const int rowA = rt * 16 + m;
    v2f sA = *(const v2f*)(mb + rowA * 2);
    v2f A;
    A.x = hi ? (ALPHA * sA.x) : sA.x;
    A.y = hi ? (-ALPHA * sA.y) : sA.y;

    float acx[8] = {}, acy[8] = {};

    v2f t = *(const v2f*)(mb + ((size_t)m) * 2);
    for (int ct = 0; ct < NTILES; ++ct) {
        const int jn = (((ct + 1) & (NTILES - 1)) << 4) + m;
        v2f tn = *(const v2f*)(mb + (size_t)jn * 2);

        v2f Bv;
        Bv.x = hi ? t.y : t.x;
        Bv.y = hi ? t.x : t.y;

        v8f c = {};
        v8f d = __builtin_amdgcn_wmma_f32_16x16x4_f32(
            false, A, false, Bv, (short)0, c, false, false);

#pragma unroll
        for (int v = 0; v < 8; ++v) {
            float att = gate_att(d[v]);
            acx[v] += att * t.x;
            acy[v] += att * t.y;
        }
        t = tn;
    }

#pragma unroll
    for (int v = 0; v < 8; ++v) {
#pragma unroll
        for (int mask = 1; mask < 16; mask <<= 1) {
            acx[v] += __shfl_xor(acx[v], mask, 32);
            acy[v] += __shfl_xor(acy[v], mask, 32);
        }
    }

    if (m == 0) {
#pragma unroll
        for (int v = 0; v < 8; ++v) {
            const int row  = rt * 16 + v + 8 * hi;
            const size_t i = (size_t)b * NN + row;
            v2f st = *(const v2f*)(sb + row * 2);
            const float cx1 = ws_ctx[i * 2];
            const float cy1 = ws_ctx[i * 2 + 1];
            // evolved = states + 0.5*(DT*ctx1 + DT*ctx2)
            float ex = st.x + 0.5f * DT * (cx1 + acx[v]);
            float ey = st.y + 0.5f * DT * (cy1 + acy[v]);
            // skip blend
            float esx = SKIP_W * st.x + (1.0f - SKIP_W) * ex;
            float esy = SKIP_W * st.y + (1.0f - SKIP_W) * ey;
            // radius adjustment (first-pass attn_sum)
            float r  = sqrtf(esx * esx + esy * esy);
            float nr = fminf(fmaxf(r + ws_sum[i] * RADIUS_RATE, R_MIN), R_MAX);
            float sc = nr / (r + 1e-8f);
            out_states[i * 2]     = esx * sc;
            out_states[i * 2 + 1] = esy * sc;
        }
    }
}

extern "C" void kernel_launch(void* const* d_in, const int* in_sizes, int n_in,
                              void* d_out, int out_size, void* d_ws, size_t ws_size,
                              hipStream_t stream) {
    const float* states = (const float*)d_in[0];

    // Output layout: new_states [B,N,2] then attention [B,N,N], flat.
    float* out_states = (float*)d_out;
    float* out_att    = (float*)d_out + (size_t)BB * NN * 2;

    // Workspace: ctx [B,N,2] | sum [B,N] | mid [B,N,2]  (~320 KB)
    float* ws_ctx = (float*)d_ws;
    float* ws_sum = ws_ctx + (size_t)BB * NN * 2;
    float* ws_mid = ws_sum + (size_t)BB * NN;

    // 1024 waves total: 128 blocks x 8 waves, one wave per 16-row tile.
    dim3 grid(BB * NN / 16 / 8);
    dim3 block(256);
    qfnn_pass1<<<grid, block, 0, stream>>>(states, out_att, ws_ctx, ws_sum, ws_mid);
    qfnn_pass2<<<grid, block, 0, stream>>>(states, ws_mid, ws_ctx, ws_sum, out_states);
}